// AdaptiveEmbeddingI2T_87746181857680
// MI455X (gfx1250) — compile-verified
//
#include <hip/hip_runtime.h>
#include <hip/hip_bf16.h>

// ---------------------------------------------------------------------------
// Problem constants (from reference)
// ---------------------------------------------------------------------------
#define B_IMG 64
#define B_CAP 64
#define T_CAP 64
#define DIM   1024
#define HDIM  128
#define T_IMG 36
#define EPS_BN 1e-5f
#define GATE_GAMMA 10.0f
#define LOG2E 1.4426950408889634f
#define GI    8   // images processed per fused block (amortizes xn loads 8x)

typedef float v2f __attribute__((ext_vector_type(2)));
typedef float v8f __attribute__((ext_vector_type(8)));

// ---- fast-math helpers (guarded; host pass falls back to libm) ------------
#if __has_builtin(__builtin_amdgcn_exp2f)
#define EXP2F(x) __builtin_amdgcn_exp2f(x)
#else
#define EXP2F(x) exp2f(x)
#endif

#if __has_builtin(__builtin_amdgcn_rsqf)
#define RSQF(x) __builtin_amdgcn_rsqf(x)
#else
#define RSQF(x) rsqrtf(x)
#endif

#if __has_builtin(__builtin_amdgcn_rcpf)
#define RCPF(x) __builtin_amdgcn_rcpf(x)
#else
#define RCPF(x) (1.0f / (x))
#endif

#if __has_builtin(__builtin_amdgcn_wmma_f32_16x16x4_f32)
#define HAVE_WMMA_F32X4 1
#else
#define HAVE_WMMA_F32X4 0
#endif

// ---------------------------------------------------------------------------
// One 16x16 fp32 output tile of C = A(row-major, lda) x B(row-major, ldb),
// accumulated over K with V_WMMA_F32_16X16X4_F32. One wave (32 lanes).
//   A fragment (16x4): lane l, vgpr j -> A[row0 + (l&15)][k0 + (l>>4)*2 + j]
//   B fragment (4x16): lane l, vgpr j -> B[k0 + (l>>4)*2 + j][col0 + (l&15)]
//   D (16x16):         lane l, vgpr r -> C[row0 + r + 8*(l>>4)][col0 + (l&15)]
// ---------------------------------------------------------------------------
__device__ __forceinline__ v8f gemm_tile_f32(const float* __restrict__ A, int lda,
                                             const float* __restrict__ B, int ldb,
                                             int K, int row0, int col0, int lane) {
  const int r  = lane & 15;
  const int hi = lane >> 4;
  v8f acc = {};
#if HAVE_WMMA_F32X4
  for (int k0 = 0; k0 < K; k0 += 4) {
    const int ka = k0 + hi * 2;
    v2f a, b;
    a.x = A[(row0 + r) * lda + ka];
    a.y = A[(row0 + r) * lda + ka + 1];
    b.x = B[ka * ldb + col0 + r];
    b.y = B[(ka + 1) * ldb + col0 + r];
    acc = __builtin_amdgcn_wmma_f32_16x16x4_f32(
        /*neg_a=*/false, a, /*neg_b=*/false, b,
        /*c_mod=*/(short)0, acc, /*reuse_a=*/false, /*reuse_b=*/false);
  }
#else
  // Scalar fallback producing identical D-layout (host pass / missing builtin).
  for (int rr = 0; rr < 8; ++rr) {
    const int row = row0 + rr + 8 * hi;
    const int col = col0 + r;
    float s = 0.f;
    for (int k = 0; k < K; ++k) s += A[row * lda + k] * B[k * ldb + col];
    acc[rr] = s;
  }
#endif
  return acc;
}

// ---------------------------------------------------------------------------
// Kernel 1: img_repr[i,d] = mean_t img_embed[i,t,d]        (64 x 1024)
// ---------------------------------------------------------------------------
__global__ void k_img_repr(const float* __restrict__ img, float* __restrict__ imgr) {
  const int idx = blockIdx.x * blockDim.x + threadIdx.x;  // 65536
  const int i = idx >> 10, d = idx & 1023;
  const float* p = img + i * (T_IMG * DIM) + d;
  float s = 0.f;
#pragma unroll
  for (int t = 0; t < T_IMG; ++t) s += p[t * DIM];
  imgr[idx] = s * (1.0f / T_IMG);
}

// ---------------------------------------------------------------------------
// Kernel 2: imgv = l2norm(img_repr) over d   (block per image row)
// ---------------------------------------------------------------------------
__global__ void k_img_norm(const float* __restrict__ imgr, float* __restrict__ imgv) {
  const int i = blockIdx.x, tid = threadIdx.x;  // 256 threads
  float s = 0.f;
#pragma unroll
  for (int c = 0; c < 4; ++c) {
    float v = imgr[i * DIM + tid + c * 256];
    s += v * v;
  }
  __shared__ float red[256];
  red[tid] = s;
  __syncthreads();
  for (int off = 128; off > 0; off >>= 1) {
    if (tid < off) red[tid] += red[tid + off];
    __syncthreads();
  }
  const float inv = RSQF(red[0]);
#pragma unroll
  for (int c = 0; c < 4; ++c) {
    const int d = tid + c * 256;
    imgv[i * DIM + d] = imgr[i * DIM + d] * inv;
  }
}

// ---------------------------------------------------------------------------
// Kernel 3: per-channel batch stats over (Bc, T_IMG): mu[d], rstd[d]
// ---------------------------------------------------------------------------
__global__ void k_stats(const float* __restrict__ cap, float* __restrict__ mu,
                        float* __restrict__ rstd) {
  const int d = blockIdx.x, tid = threadIdx.x;  // 128 threads, d in [0,1024)
  float s = 0.f, s2 = 0.f;
  for (int sidx = tid; sidx < B_CAP * T_IMG; sidx += 128) {
    const int b = sidx / T_IMG, t = sidx - b * T_IMG;
    const float v = cap[(b * T_CAP + t) * DIM + d];
    s += v;
    s2 += v * v;
  }
  __shared__ float rs[128], rq[128];
  rs[tid] = s;
  rq[tid] = s2;
  __syncthreads();
  for (int off = 64; off > 0; off >>= 1) {
    if (tid < off) { rs[tid] += rs[tid + off]; rq[tid] += rq[tid + off]; }
    __syncthreads();
  }
  if (tid == 0) {
    const float inv_n = 1.0f / (B_CAP * T_IMG);
    const float m = rs[0] * inv_n;
    const float var = rq[0] * inv_n - m * m;
    mu[d] = m;
    rstd[d] = RSQF(var + EPS_BN);
  }
}

// ---------------------------------------------------------------------------
// Kernel 4 (WMMA): H = relu(img_repr @ W1 + b1)   (64x1024)x(1024x128)
// blockIdx.y selects gamma/beta branch. One wave per block, one 16x16 tile.
// ---------------------------------------------------------------------------
__global__ void k_gemm1(const float* __restrict__ imgr,
                        const float* __restrict__ Wg1, const float* __restrict__ bg1,
                        const float* __restrict__ Wb1, const float* __restrict__ bb1,
                        float* __restrict__ Hg, float* __restrict__ Hb) {
  const int lane = threadIdx.x;      // 32 threads = 1 wave
  const int tile = blockIdx.x;       // 0..31 : 4 M-tiles x 8 N-tiles
  const int tm = tile >> 3, tn = tile & 7;
  const bool isB = (blockIdx.y != 0);
  const float* W = isB ? Wb1 : Wg1;
  const float* bias = isB ? bb1 : bg1;
  float* H = isB ? Hb : Hg;

  v8f acc = gemm_tile_f32(imgr, DIM, W, HDIM, DIM, tm * 16, tn * 16, lane);

  const int r = lane & 15, hi = lane >> 4;
#pragma unroll
  for (int rr = 0; rr < 8; ++rr) {
    const int M = tm * 16 + rr + 8 * hi;
    const int N = tn * 16 + r;
    H[M * HDIM + N] = fmaxf(acc[rr] + bias[N], 0.0f);
  }
}

// ---------------------------------------------------------------------------
// Kernel 5 (WMMA): gam = 1 + H @ W2 + b2 ; bet = H @ W2 + b2
//                  (64x128)x(128x1024)
// ---------------------------------------------------------------------------
__global__ void k_gemm2(const float* __restrict__ Hg, const float* __restrict__ Hb,
                        const float* __restrict__ Wg2, const float* __restrict__ bg2,
                        const float* __restrict__ Wb2, const float* __restrict__ bb2,
                        float* __restrict__ gam, float* __restrict__ bet) {
  const int lane = threadIdx.x;      // 32 threads = 1 wave
  const int tile = blockIdx.x;       // 0..255 : 4 M-tiles x 64 N-tiles
  const int tm = tile >> 6, tn = tile & 63;
  const bool isB = (blockIdx.y != 0);
  const float* H = isB ? Hb : Hg;
  const float* W = isB ? Wb2 : Wg2;
  const float* bias = isB ? bb2 : bg2;
  float* Out = isB ? bet : gam;
  const float add = isB ? 0.0f : 1.0f;

  v8f acc = gemm_tile_f32(H, HDIM, W, DIM, HDIM, tm * 16, tn * 16, lane);

  const int r = lane & 15, hi = lane >> 4;
#pragma unroll
  for (int rr = 0; rr < 8; ++rr) {
    const int M = tm * 16 + rr + 8 * hi;
    const int N = tn * 16 + r;
    Out[M * DIM + N] = acc[rr] + bias[N] + add;
  }
}

// ---------------------------------------------------------------------------
// Kernel 6: fused BN-apply + gate/pool/norm/dot.
// Block = (i-group gi0 = blockIdx.x*GI, caption b = blockIdx.y), 256 threads.
// Each thread owns 4 channels d; it normalizes and keeps the 36 timesteps of
// its channel in REGISTERS once, then runs GI=8 per-image softmax gates from
// registers (8x less L2 traffic than one image per block; exp-bound after).
//   softmax shift-invariance: beta drops from the gate; sum_t mask = 1, so
//   txt_vec = (gam*S + bet)/36 with S = sum_t xn*e / sum_t e,
//   e = exp2(a2*xn - m2), a2 = 10*log2(e)*gam, m2 = max(a2*xmax, a2*xmin).
// Block-reduce per image: dot = sum_d imgv*tv, nrm = sum_d tv^2;
// sims[i,b] = dot * rsqrt(nrm).
// ---------------------------------------------------------------------------
__global__ void k_fuse(const float* __restrict__ cap, const float* __restrict__ mu,
                       const float* __restrict__ rstd, const float* __restrict__ gam,
                       const float* __restrict__ bet, const float* __restrict__ imgv,
                       float* __restrict__ out) {
  const int i0 = blockIdx.x * GI;
  const int b = blockIdx.y;
  const int tid = threadIdx.x;
  const float* capb = cap + (size_t)b * (T_CAP * DIM);

  float dot[GI], nrm[GI];
#pragma unroll
  for (int gi = 0; gi < GI; ++gi) { dot[gi] = 0.f; nrm[gi] = 0.f; }

#pragma unroll 1
  for (int c = 0; c < 4; ++c) {
    const int d = tid + c * 256;
    const float mu_d = mu[d];
    const float rs_d = rstd[d];

    // Normalize once into registers; track extrema for the softmax max trick.
    float x[T_IMG];
    float mx = -3.0e38f, mn = 3.0e38f;
#pragma unroll
    for (int t = 0; t < T_IMG; ++t) {
      x[t] = (capb[t * DIM + d] - mu_d) * rs_d;
      mx = fmaxf(mx, x[t]);
      mn = fminf(mn, x[t]);
    }

#pragma unroll
    for (int gi = 0; gi < GI; ++gi) {
      const int i = i0 + gi;
      const float g = gam[i * DIM + d];
      const float be = bet[i * DIM + d];
      const float a2 = (GATE_GAMMA * LOG2E) * g;   // exp -> exp2 scale
      const float m2 = fmaxf(a2 * mx, a2 * mn);    // exact max_t of a2*x
      float se = 0.f, sx = 0.f;
#pragma unroll
      for (int t = 0; t < T_IMG; ++t) {
        const float e = EXP2F(a2 * x[t] - m2);
        se += e;
        sx += x[t] * e;
      }
      const float S = sx * RCPF(se);
      const float tv = (g * S + be) * (1.0f / T_IMG);
      dot[gi] += imgv[i * DIM + d] * tv;
      nrm[gi] += tv * tv;
    }
  }

  __shared__ float rdot[256], rnrm[256];
#pragma unroll 1
  for (int gi = 0; gi < GI; ++gi) {
    __syncthreads();            // protect LDS reuse across gi iterations
    rdot[tid] = dot[gi];
    rnrm[tid] = nrm[gi];
    __syncthreads();
    for (int off = 128; off > 0; off >>= 1) {
      if (tid < off) {
        rdot[tid] += rdot[tid + off];
        rnrm[tid] += rnrm[tid + off];
      }
      __syncthreads();
    }
    if (tid == 0) out[(i0 + gi) * B_CAP + b] = rdot[0] * RSQF(rnrm[0]);
  }
}

// ---------------------------------------------------------------------------
// Launch
// ---------------------------------------------------------------------------
extern "C" void kernel_launch(void* const* d_in, const int* in_sizes, int n_in,
                              void* d_out, int out_size, void* d_ws, size_t ws_size,
                              hipStream_t stream) {
  const float* img = (const float*)d_in[0];   // (64,36,1024)
  const float* cap = (const float*)d_in[1];   // (64,64,1024)
  // d_in[2] = lens (unused by reference math)
  const float* Wg1 = (const float*)d_in[3];
  const float* bg1 = (const float*)d_in[4];
  const float* Wg2 = (const float*)d_in[5];
  const float* bg2 = (const float*)d_in[6];
  const float* Wb1 = (const float*)d_in[7];
  const float* bb1 = (const float*)d_in[8];
  const float* Wb2 = (const float*)d_in[9];
  const float* bb2 = (const float*)d_in[10];
  float* out = (float*)d_out;                 // (64,64) sims[i,b]

  float* ws = (float*)d_ws;
  size_t off = 0;
  float* imgr = ws + off; off += (size_t)B_IMG * DIM;   // 65536
  float* imgv = ws + off; off += (size_t)B_IMG * DIM;   // 65536
  float* mu   = ws + off; off += DIM;                   // 1024
  float* rstd = ws + off; off += DIM;                   // 1024
  float* Hg   = ws + off; off += (size_t)B_IMG * HDIM;  // 8192
  float* Hb   = ws + off; off += (size_t)B_IMG * HDIM;  // 8192
  float* gam  = ws + off; off += (size_t)B_IMG * DIM;   // 65536
  float* bet  = ws + off; off += (size_t)B_IMG * DIM;   // 65536
  (void)in_sizes; (void)n_in; (void)out_size; (void)ws_size;

  k_img_repr<<<(B_IMG * DIM) / 256, 256, 0, stream>>>(img, imgr);
  k_img_norm<<<B_IMG, 256, 0, stream>>>(imgr, imgv);
  k_stats<<<DIM, 128, 0, stream>>>(cap, mu, rstd);
  k_gemm1<<<dim3(32, 2), 32, 0, stream>>>(imgr, Wg1, bg1, Wb1, bb1, Hg, Hb);
  k_gemm2<<<dim3(256, 2), 32, 0, stream>>>(Hg, Hb, Wg2, bg2, Wb2, bb2, gam, bet);
  k_fuse<<<dim3(B_IMG / GI, B_CAP), 256, 0, stream>>>(cap, mu, rstd, gam, bet, imgv, out);
}